// Attention_18021682774359
// MI455X (gfx1250) — compile-verified
//
#include <hip/hip_runtime.h>
#include <hip/hip_bf16.h>

// ---------------- problem constants ----------------
constexpr int B   = 64;
constexpr int CL  = 400;   // 25 * 16
constexpr int QL  = 50;
constexpr int QLP = 64;    // QL padded to multiple of 16
constexpr int H   = 1024;  // 64 * 16
constexpr float NEG_INF = -1e30f;

typedef __attribute__((ext_vector_type(2))) float v2f;
typedef __attribute__((ext_vector_type(8))) float v8f;

// D = A(16x4) * B(4x16) + C, full fp32 (CDNA5 V_WMMA_F32_16X16X4_F32)
static __device__ __forceinline__ v8f wmma_f32_4(v2f a, v2f b, v8f c) {
  return __builtin_amdgcn_wmma_f32_16x16x4_f32(
      false, a, false, b, (short)0, c, false, false);
}

static __device__ __forceinline__ float wave_red_max(float v) {
  for (int m = 16; m >= 1; m >>= 1) v = fmaxf(v, __shfl_xor(v, m, 32));
  return v;
}
static __device__ __forceinline__ float wave_red_sum(float v) {
  for (int m = 16; m >= 1; m >>= 1) v += __shfl_xor(v, m, 32);
  return v;
}

// ---------------- K0: build padded q tensors --------------------------
// q_pad[b, jp, h]  = jp<QL ? q[b,jp,h]          : 0   (for final a = s1@q)
// qc_pad[b, jp, h] = jp<QL ? q[b,jp,h]*cqw[h]   : 0   (for similarity GEMM)
// one block per (b, jp) row
__global__ void padq_kernel(const float* __restrict__ q,
                            const float* __restrict__ cqw,
                            float* __restrict__ q_pad,
                            float* __restrict__ qc_pad) {
  int b  = blockIdx.x / QLP;
  int jp = blockIdx.x % QLP;
  const float* src = q + ((size_t)b * QL + jp) * H;   // only deref if jp<QL
  float* d0 = q_pad  + ((size_t)b * QLP + jp) * H;
  float* d1 = qc_pad + ((size_t)b * QLP + jp) * H;
  bool valid = jp < QL;
  for (int h = threadIdx.x; h < H; h += blockDim.x) {
    float v = valid ? src[h] : 0.f;
    d0[h] = v;
    d1[h] = v * cqw[h];
  }
}

// ---------------- K1: per-row dot with a weight vector ----------------
__global__ void rowdot_kernel(const float* __restrict__ x,
                              const float* __restrict__ w,
                              float* __restrict__ out, int nrows) {
  int wave = threadIdx.x >> 5, lane = threadIdx.x & 31;
  int row = blockIdx.x * 8 + wave;
  if (row >= nrows) return;
  const float* xr = x + (size_t)row * H;
  float s = 0.f;
  for (int h = lane; h < H; h += 32) s += xr[h] * w[h];
  s = wave_red_sum(s);
  if (lane == 0) out[row] = s;
}

// ---------------- K2: similarity S[b,i,jp] ----------------------------
// One wave per (b, i-tile of 16); the wave carries ALL four 16x16 j-tiles
// (4 accumulators) so each c row is loaded exactly once from global.
// S = WMMA_{K=H}( c , (q .* cqw)^T ) + cw_i + qw_j + bias
__global__ void sim_kernel(const float* __restrict__ c,
                           const float* __restrict__ qc_pad,
                           const float* __restrict__ cw,
                           const float* __restrict__ qw,
                           const float* __restrict__ biasp,
                           float* __restrict__ S) {
  int wg = blockIdx.x * 8 + (threadIdx.x >> 5);   // wave-tile id
  int lane = threadIdx.x & 31;
  int b  = wg / (CL / 16);
  int i0 = (wg % (CL / 16)) * 16;
  int m  = lane & 15;        // A row / B col within tile
  int hi = lane >> 4;        // selects K pair {0,1} vs {2,3}

  const float* crow = c + ((size_t)b * CL + i0 + m) * H;
  const float* q0 = qc_pad + ((size_t)b * QLP +  0 + m) * H;
  const float* q1 = qc_pad + ((size_t)b * QLP + 16 + m) * H;
  const float* q2 = qc_pad + ((size_t)b * QLP + 32 + m) * H;
  const float* q3 = qc_pad + ((size_t)b * QLP + 48 + m) * H;

  v8f acc0 = {0.f,0.f,0.f,0.f,0.f,0.f,0.f,0.f};
  v8f acc1 = acc0, acc2 = acc0, acc3 = acc0;

  for (int k0 = 0; k0 < H; k0 += 4) {
    int ka = k0 + hi * 2;
    if ((k0 & 63) == 0) __builtin_prefetch(crow + k0 + 64, 0, 1);
    v2f A, B0, B1, B2, B3;
    A.x  = crow[ka]; A.y  = crow[ka + 1];
    B0.x = q0[ka];   B0.y = q0[ka + 1];
    B1.x = q1[ka];   B1.y = q1[ka + 1];
    B2.x = q2[ka];   B2.y = q2[ka + 1];
    B3.x = q3[ka];   B3.y = q3[ka + 1];
    acc0 = wmma_f32_4(A, B0, acc0);
    acc1 = wmma_f32_4(A, B1, acc1);
    acc2 = wmma_f32_4(A, B2, acc2);
    acc3 = wmma_f32_4(A, B3, acc3);
  }

  float bs = biasp[0];
  #pragma unroll
  for (int t = 0; t < 4; ++t) {
    v8f acc = (t == 0) ? acc0 : (t == 1) ? acc1 : (t == 2) ? acc2 : acc3;
    int jn = t * 16 + m;                      // output column (lane-mapped n)
    float qwv = (jn < QL) ? qw[b * QL + jn] : 0.f;
    #pragma unroll
    for (int r = 0; r < 8; ++r) {
      int mi = r + 8 * hi;                    // C/D layout: M = r + 8*(lane>=16)
      float s = acc[r] + cw[b * CL + i0 + mi] + qwv + bs;
      S[((size_t)b * CL + i0 + mi) * QLP + jn] = s;
    }
  }
}

// ---------------- K3: s1 = softmax over j (row-wise), zero-padded ------
__global__ void softmax_row_kernel(const float* __restrict__ S,
                                   const int* __restrict__ q_mask,
                                   float* __restrict__ s1) {
  int wave = threadIdx.x >> 5, lane = threadIdx.x & 31;
  int row = blockIdx.x * 8 + wave;            // row = b*CL + i
  int b = row / CL;
  const float* Sr = S + (size_t)row * QLP;

  int j1 = lane + 32;
  bool e1 = j1 < QL;
  float v0 = (q_mask[b * QL + lane] != 0) ? Sr[lane] : NEG_INF;
  float v1 = e1 ? ((q_mask[b * QL + j1] != 0) ? Sr[j1] : NEG_INF) : -INFINITY;

  float mx = wave_red_max(fmaxf(v0, v1));
  float e0 = __expf(v0 - mx);
  float ee1 = e1 ? __expf(v1 - mx) : 0.f;
  float sum = wave_red_sum(e0 + ee1);
  float inv = 1.f / sum;

  float* o = s1 + (size_t)row * QLP;
  o[lane] = e0 * inv;
  o[j1]   = e1 ? ee1 * inv : 0.f;
}

// ---------------- K4: s2 = softmax over i (column-wise), pad cols = 0 --
__global__ void softmax_col_kernel(const float* __restrict__ S,
                                   const int* __restrict__ c_mask,
                                   float* __restrict__ s2) {
  __shared__ float red[128];
  int b = blockIdx.x / QLP;
  int j = blockIdx.x % QLP;
  int t = threadIdx.x;

  if (j >= QL) {
    for (int i = t; i < CL; i += 128)
      s2[((size_t)b * CL + i) * QLP + j] = 0.f;
    return;
  }

  float vals[4];
  float lmax = -INFINITY;
  #pragma unroll
  for (int it = 0; it < 4; ++it) {
    int i = t + it * 128;
    float v = -INFINITY;
    if (i < CL)
      v = (c_mask[b * CL + i] != 0) ? S[((size_t)b * CL + i) * QLP + j]
                                    : NEG_INF;
    vals[it] = v;
    lmax = fmaxf(lmax, v);
  }
  red[t] = lmax; __syncthreads();
  for (int s = 64; s >= 1; s >>= 1) {
    if (t < s) red[t] = fmaxf(red[t], red[t + s]);
    __syncthreads();
  }
  float mx = red[0]; __syncthreads();

  float lsum = 0.f;
  #pragma unroll
  for (int it = 0; it < 4; ++it) {
    int i = t + it * 128;
    float e = (i < CL) ? __expf(vals[it] - mx) : 0.f;
    vals[it] = e;
    lsum += e;
  }
  red[t] = lsum; __syncthreads();
  for (int s = 64; s >= 1; s >>= 1) {
    if (t < s) red[t] += red[t + s];
    __syncthreads();
  }
  float inv = 1.f / red[0];

  #pragma unroll
  for (int it = 0; it < 4; ++it) {
    int i = t + it * 128;
    if (i < CL) s2[((size_t)b * CL + i) * QLP + j] = vals[it] * inv;
  }
}

// ---------------- K5: s2tc[b, j, h] = sum_i s2[b,i,j] * c[b,i,h] -------
// WMMA GEMM: M = QLP (rows j>=QL are zero since s2 pad cols are 0),
// K = CL, N = H. One wave per 16x16 (j,h) tile.
__global__ void s2tc_kernel(const float* __restrict__ s2,
                            const float* __restrict__ c,
                            float* __restrict__ s2tc) {
  int wg = blockIdx.x * 8 + (threadIdx.x >> 5);
  int lane = threadIdx.x & 31;
  int b   = wg / ((QLP / 16) * (H / 16));
  int rem = wg % ((QLP / 16) * (H / 16));
  int j0  = (rem / (H / 16)) * 16;
  int h0  = (rem % (H / 16)) * 16;
  int m  = lane & 15;
  int hi = lane >> 4;
  int j  = j0 + m;                                  // A row
  int n  = h0 + m;                                  // B col

  v8f acc = {0.f,0.f,0.f,0.f,0.f,0.f,0.f,0.f};
  for (int k0 = 0; k0 < CL; k0 += 4) {
    int ka = k0 + hi * 2;
    if ((k0 & 63) == 0)
      __builtin_prefetch(c + ((size_t)b * CL + k0 + 64) * H + n, 0, 1);
    v2f A, Bv;
    A.x = s2[((size_t)b * CL + ka)     * QLP + j];  // A[m][k] = s2[b,k,j]
    A.y = s2[((size_t)b * CL + ka + 1) * QLP + j];
    Bv.x = c[((size_t)b * CL + ka)     * H + n];    // B[k][n] = c[b,k,h]
    Bv.y = c[((size_t)b * CL + ka + 1) * H + n];
    acc = wmma_f32_4(A, Bv, acc);
  }
  #pragma unroll
  for (int r = 0; r < 8; ++r) {
    int jo = j0 + r + 8 * hi;
    s2tc[((size_t)b * QLP + jo) * H + h0 + (lane & 15)] = acc[r];
  }
}

// ---------------- K6: a = s1@q_pad, bb = s1@s2tc; write [c,a,c*a,c*bb] --
// One wave per 16x16 (i,h) tile; K = QLP = 64, all operands zero-padded.
__global__ void final_kernel(const float* __restrict__ s1,
                             const float* __restrict__ q_pad,
                             const float* __restrict__ s2tc,
                             const float* __restrict__ c,
                             float* __restrict__ out) {
  int wg = blockIdx.x * 8 + (threadIdx.x >> 5);
  int lane = threadIdx.x & 31;
  int b   = wg / ((CL / 16) * (H / 16));
  int rem = wg % ((CL / 16) * (H / 16));
  int i0  = (rem / (H / 16)) * 16;
  int h0  = (rem % (H / 16)) * 16;
  int m  = lane & 15;
  int hi = lane >> 4;
  int i  = i0 + m;                                  // A row
  int n  = h0 + m;                                  // B col

  const float* Ar = s1 + ((size_t)b * CL + i) * QLP;
  const float* Bq = q_pad + (size_t)b * QLP * H + n;
  const float* Bt = s2tc  + (size_t)b * QLP * H + n;

  v8f acca = {0.f,0.f,0.f,0.f,0.f,0.f,0.f,0.f};
  v8f accb = acca;
  for (int k0 = 0; k0 < QLP; k0 += 4) {
    int ka = k0 + hi * 2;
    v2f A, Bqv, Btv;
    A.x = Ar[ka];
    A.y = Ar[ka + 1];
    Bqv.x = Bq[(size_t)ka * H];
    Bqv.y = Bq[(size_t)(ka + 1) * H];
    Btv.x = Bt[(size_t)ka * H];
    Btv.y = Bt[(size_t)(ka + 1) * H];
    acca = wmma_f32_4(A, Bqv, acca);
    accb = wmma_f32_4(A, Btv, accb);
  }
  #pragma unroll
  for (int r = 0; r < 8; ++r) {
    int io = i0 + r + 8 * hi;
    int ho = h0 + (lane & 15);
    float cv = c[((size_t)b * CL + io) * H + ho];
    float av = acca[r];
    float bv = accb[r];
    size_t orow = ((size_t)b * CL + io) * (size_t)(4 * H);
    out[orow + ho]          = cv;
    out[orow + H + ho]      = av;
    out[orow + 2 * H + ho]  = cv * av;
    out[orow + 3 * H + ho]  = cv * bv;
  }
}

extern "C" void kernel_launch(void* const* d_in, const int* in_sizes, int n_in,
                              void* d_out, int out_size, void* d_ws, size_t ws_size,
                              hipStream_t stream) {
  const float* c        = (const float*)d_in[0];
  const float* q        = (const float*)d_in[1];
  const int*   c_mask   = (const int*)d_in[2];
  const int*   q_mask   = (const int*)d_in[3];
  const float* c_weight = (const float*)d_in[4];
  const float* q_weight = (const float*)d_in[5];
  const float* cq_weight= (const float*)d_in[6];
  const float* bias     = (const float*)d_in[7];
  float* out = (float*)d_out;

  // workspace (floats): cw | qw | S | s1 | s2 | s2tc | q_pad | qc_pad (~70 MB)
  float* cw     = (float*)d_ws;
  float* qw     = cw   + (size_t)B * CL;
  float* S      = qw   + (size_t)B * QL;
  float* s1     = S    + (size_t)B * CL * QLP;
  float* s2     = s1   + (size_t)B * CL * QLP;
  float* s2tc   = s2   + (size_t)B * CL * QLP;
  float* q_pad  = s2tc + (size_t)B * QLP * H;
  float* qc_pad = q_pad + (size_t)B * QLP * H;

  padq_kernel<<<B * QLP, 256, 0, stream>>>(q, cq_weight, q_pad, qc_pad);
  rowdot_kernel<<<(B * CL + 7) / 8, 256, 0, stream>>>(c, c_weight, cw, B * CL);
  rowdot_kernel<<<(B * QL + 7) / 8, 256, 0, stream>>>(q, q_weight, qw, B * QL);
  sim_kernel<<<(B * (CL / 16)) / 8, 256, 0, stream>>>(c, qc_pad, cw, qw, bias, S);
  softmax_row_kernel<<<(B * CL) / 8, 256, 0, stream>>>(S, q_mask, s1);
  softmax_col_kernel<<<B * QLP, 128, 0, stream>>>(S, c_mask, s2);
  s2tc_kernel<<<(B * (QLP / 16) * (H / 16)) / 8, 256, 0, stream>>>(s2, c, s2tc);
  final_kernel<<<(B * (CL / 16) * (H / 16)) / 8, 256, 0, stream>>>(s1, q_pad, s2tc, c, out);
}